// EALayer_6416681140993
// MI455X (gfx1250) — compile-verified
//
#include <hip/hip_runtime.h>
#include <hip/hip_bf16.h>

typedef float v2f __attribute__((ext_vector_type(2)));
typedef float v8f __attribute__((ext_vector_type(8)));

#define E_HID 128

// ---------------------------------------------------------------------------
// Kernel 1: init — zero x_e region of d_out, reset softmax scalars in ws
// ---------------------------------------------------------------------------
__global__ void init_kernel(float* __restrict__ xe, int n_xe,
                            int* __restrict__ gmax_bits, float* __restrict__ gsum) {
    long i = (long)blockIdx.x * blockDim.x + threadIdx.x;
    if (i < n_xe) xe[i] = 0.0f;
    if (i == 0) {
        *gmax_bits = (int)0xFF800000u;  // bits of -inf (monotone-map init)
        *gsum = 0.0f;
    }
}

// ---------------------------------------------------------------------------
// Kernel 2/3: D[M x 128] = A[M x 128] * W^T, W is [128 x 128] row-major.
// One wave (32 lanes) per 16x16 output tile, V_WMMA_F32_16X16X4_F32,
// K=128 fully unrolled (32 WMMAs), branch-free inner loop (EXEC all-ones):
//  - ragged M handled by CLAMPING the A row index (lane m's A row only feeds
//    output row m, which we simply don't store when out of range)
//  - A/B fragments are contiguous 8B-aligned float2 -> global_load_b64
// Layouts per ISA:
//  A frag (16x4): lane L (m=L&15, half=L>>4) holds A[m][k+2*half +{0,1}].
//  B frag (4x16): lane L holds W[n=L&15 (+16*nT)][k+2*half +{0,1}].
//  C/D (16x16):  acc[v] -> row = 16*mT + v + 8*half, col = 16*nT + (L&15).
// ---------------------------------------------------------------------------
__global__ __launch_bounds__(32)
void wmma_gemm_abt(const float* __restrict__ A,
                   const float* __restrict__ W,
                   float* __restrict__ D,
                   int M) {
    const int lane = threadIdx.x & 31;
    const int half = lane >> 4;
    const int lm   = lane & 15;
    const int nT   = blockIdx.x;       // 128/16 = 8 tiles in N
    const int mT   = blockIdx.y;       // ceil(M/16) tiles in M

    const int arow   = mT * 16 + lm;
    const int arow_c = (arow < M) ? arow : (M - 1);   // clamp ragged tile
    const int wrow   = nT * 16 + lm;

    const v2f* __restrict__ Afrag = (const v2f*)(A + (long)arow_c * E_HID + 2 * half);
    const v2f* __restrict__ Wfrag = (const v2f*)(W + (long)wrow   * E_HID + 2 * half);

    v8f acc = {};
#pragma unroll
    for (int k = 0; k < E_HID; k += 4) {
        v2f a = Afrag[k >> 1];   // A[row][k+2*half], A[row][k+2*half+1]
        v2f b = Wfrag[k >> 1];   // W[col][k+2*half], W[col][k+2*half+1]
        acc = __builtin_amdgcn_wmma_f32_16x16x4_f32(
            /*neg_a=*/false, a, /*neg_b=*/false, b,
            /*c_mod=*/(short)0, acc, /*reuse_a=*/false, /*reuse_b=*/false);
    }
#pragma unroll
    for (int v = 0; v < 8; ++v) {
        const int r = mT * 16 + v + 8 * half;
        const int c = nT * 16 + lm;
        if (r < M) D[(long)r * E_HID + c] = acc[v];
    }
}

// ---------------------------------------------------------------------------
// Kernel 4: one wave32 per edge: dp[e] = <x[src]+r_emb[et], x[dst]>,
// res_att passthrough, block max -> global atomic float-max.
// ---------------------------------------------------------------------------
__device__ __forceinline__ void atomic_max_float(int* addr_bits, float val) {
    if (val >= 0.0f) atomicMax(addr_bits, __float_as_int(val));
    else             atomicMin((unsigned int*)addr_bits, __float_as_uint(val));
}

__global__ void edge_dot_kernel(const float* __restrict__ x,
                                const int* __restrict__ src,
                                const int* __restrict__ dst,
                                const int* __restrict__ etype,
                                const float* __restrict__ r_emb,
                                const float* __restrict__ res_att,
                                float* __restrict__ dp,
                                float* __restrict__ out_res,
                                int* __restrict__ gmax_bits, int E) {
    __shared__ float smax[8];
    const int wid  = threadIdx.x >> 5;
    const int lane = threadIdx.x & 31;
    const long e   = (long)blockIdx.x * 8 + wid;

    float d = -__builtin_huge_valf();
    if (e < E) {
        const int s = src[e], t = dst[e], r = etype[e];
        const float4 h  = ((const float4*)(x     + (long)s * E_HID))[lane];
        const float4 re = ((const float4*)(r_emb + (long)r * E_HID))[lane];
        const float4 tl = ((const float4*)(x     + (long)t * E_HID))[lane];
        float p = (h.x + re.x) * tl.x + (h.y + re.y) * tl.y
                + (h.z + re.z) * tl.z + (h.w + re.w) * tl.w;
        p += __shfl_xor(p, 16, 32);
        p += __shfl_xor(p,  8, 32);
        p += __shfl_xor(p,  4, 32);
        p += __shfl_xor(p,  2, 32);
        p += __shfl_xor(p,  1, 32);
        if (lane == 0) { dp[e] = p; out_res[e] = res_att[e]; }
        d = p;
    }
    if (lane == 0) smax[wid] = d;
    __syncthreads();
    if (threadIdx.x == 0) {
        float m = smax[0];
#pragma unroll
        for (int i = 1; i < 8; ++i) m = fmaxf(m, smax[i]);
        atomic_max_float(gmax_bits, m);
    }
}

// ---------------------------------------------------------------------------
// Kernel 5: expv[e] = exp(dp[e]-gmax); block sum -> global atomic add.
// ---------------------------------------------------------------------------
__global__ void exp_sum_kernel(const float* __restrict__ dp,
                               float* __restrict__ expv,
                               const int* __restrict__ gmax_bits,
                               float* __restrict__ gsum, int E) {
    __shared__ float ssum[256];
    const float gmax = __int_as_float(*gmax_bits);
    const long i = (long)blockIdx.x * 256 + threadIdx.x;
    float v = 0.0f;
    if (i < E) {
        v = __expf(dp[i] - gmax);
        expv[i] = v;
    }
    ssum[threadIdx.x] = v;
    __syncthreads();
    for (int s = 128; s > 0; s >>= 1) {
        if (threadIdx.x < s) ssum[threadIdx.x] += ssum[threadIdx.x + s];
        __syncthreads();
    }
    if (threadIdx.x == 0) atomicAdd(gsum, ssum[0]);
}

// ---------------------------------------------------------------------------
// Kernel 6: one wave32 per edge: x_e[dst] += attn * (x[src]+r_emb[et]).
// Gathers hit L2 (x fits in 192MB L2); scatter via global_atomic_add_f32.
// ---------------------------------------------------------------------------
__global__ void scatter_kernel(const float* __restrict__ x,
                               const int* __restrict__ src,
                               const int* __restrict__ dst,
                               const int* __restrict__ etype,
                               const float* __restrict__ r_emb,
                               const float* __restrict__ expv,
                               const float* __restrict__ gsum,
                               float* __restrict__ xe, int E) {
    const int wid  = threadIdx.x >> 5;
    const int lane = threadIdx.x & 31;
    const long e   = (long)blockIdx.x * 8 + wid;
    if (e >= E) return;
    const float coef = expv[e] / *gsum;
    const int s = src[e], t = dst[e], r = etype[e];
    const float4 h  = ((const float4*)(x     + (long)s * E_HID))[lane];
    const float4 re = ((const float4*)(r_emb + (long)r * E_HID))[lane];
    float* o = xe + (long)t * E_HID + lane * 4;
    atomicAdd(o + 0, coef * (h.x + re.x));
    atomicAdd(o + 1, coef * (h.y + re.y));
    atomicAdd(o + 2, coef * (h.z + re.z));
    atomicAdd(o + 3, coef * (h.w + re.w));
}

// ---------------------------------------------------------------------------
// Kernel 7: ReLU over x_e.
// ---------------------------------------------------------------------------
__global__ void relu_kernel(float* __restrict__ xe, int n) {
    long i = (long)blockIdx.x * blockDim.x + threadIdx.x;
    if (i < n) xe[i] = fmaxf(xe[i], 0.0f);
}

// ---------------------------------------------------------------------------
extern "C" void kernel_launch(void* const* d_in, const int* in_sizes, int n_in,
                              void* d_out, int out_size, void* d_ws, size_t ws_size,
                              hipStream_t stream) {
    const float* x        = (const float*)d_in[0];
    const int*   ei       = (const int*)  d_in[1];   // [2, E]
    const int*   etype    = (const int*)  d_in[2];   // [E]
    const float* rel_emb  = (const float*)d_in[3];   // [R, 128]
    const float* res_att  = (const float*)d_in[4];   // [E]
    const float* ww_w     = (const float*)d_in[5];   // [128, 128]
    const float* rel_w    = (const float*)d_in[6];   // [128, 128]

    const int n_nodes = in_sizes[0] / E_HID;         // 50000
    const int E       = in_sizes[2];                 // 600000
    const int R       = in_sizes[3] / E_HID;         // 500

    const int* src = ei;
    const int* dst = ei + E;

    // output layout: x_e | rel_out | res_att
    float* out_xe  = (float*)d_out;
    float* out_rel = out_xe + (long)n_nodes * E_HID;
    float* out_res = out_rel + (long)R * E_HID;

    // workspace layout
    char*  ws        = (char*)d_ws;
    int*   gmax_bits = (int*)  (ws + 0);
    float* gsum      = (float*)(ws + 128);
    float* r_emb     = (float*)(ws + 256);                           // R*128 f32
    float* dp        = (float*)(ws + 256 + (size_t)R * E_HID * 4);   // E f32
    float* expv      = dp + E;                                       // E f32

    const int n_xe = n_nodes * E_HID;

    // 1) zero x_e + reset softmax scalars
    init_kernel<<<(n_xe + 255) / 256, 256, 0, stream>>>(out_xe, n_xe, gmax_bits, gsum);

    // 2) r_emb = rel_emb @ ww_w.T   (WMMA f32)
    {
        dim3 grid(E_HID / 16, (R + 15) / 16);
        wmma_gemm_abt<<<grid, 32, 0, stream>>>(rel_emb, ww_w, r_emb, R);
    }
    // 3) rel_out = rel_emb @ rel_w.T (WMMA f32) -> directly into d_out
    {
        dim3 grid(E_HID / 16, (R + 15) / 16);
        wmma_gemm_abt<<<grid, 32, 0, stream>>>(rel_emb, rel_w, out_rel, R);
    }

    // 4) per-edge dot products + global max + res_att passthrough
    edge_dot_kernel<<<(E + 7) / 8, 256, 0, stream>>>(
        x, src, dst, etype, r_emb, res_att, dp, out_res, gmax_bits, E);

    // 5) exp + global sum
    exp_sum_kernel<<<(E + 255) / 256, 256, 0, stream>>>(dp, expv, gmax_bits, gsum, E);

    // 6) weighted scatter-add
    scatter_kernel<<<(E + 7) / 8, 256, 0, stream>>>(
        x, src, dst, etype, r_emb, expv, gsum, out_xe, E);

    // 7) ReLU
    relu_kernel<<<(n_xe + 255) / 256, 256, 0, stream>>>(out_xe, n_xe);
}